// GATNEModel_50929722196074
// MI455X (gfx1250) — compile-verified
//
#include <hip/hip_runtime.h>
#include <hip/hip_bf16.h>
#include <math.h>
#include <stdint.h>

// GATNE forward for MI455X (gfx1250, wave32).
#define TT 8      // edge types
#define UU 64     // embedding_u_size
#define EE 512    // embedding size
#define AA 32     // dim_a
#define NBB 10    // neighbors per type

typedef __attribute__((ext_vector_type(2))) float v2f;
typedef __attribute__((ext_vector_type(8))) float v8f;

// ---------------------------------------------------------------------------
// Kernel 1: per-sample neighbor gather + type attention -> agg[B][64]
// One block per sample b; wave t (of 8) owns edge-type t.
// Gathered rows are 256B contiguous -> two coalesced 128B reads per wave.
// ---------------------------------------------------------------------------
__global__ __launch_bounds__(256) void gatne_gather_attn(
    const float* __restrict__ node_type_emb,   // [N, T, U]
    const float* __restrict__ ws1,             // [T, U, A]
    const float* __restrict__ ws2,             // [T, A]
    const int*   __restrict__ train_types,     // [B]
    const int*   __restrict__ node_neigh,      // [B, T, NB]
    float*       __restrict__ agg_out)         // [B, U]
{
    __shared__ float s_nte[TT][UU];
    __shared__ float s_score[TT];

    const int b    = blockIdx.x;
    const int tid  = threadIdx.x;
    const int wv   = tid >> 5;     // wave id == edge type
    const int lane = tid & 31;
    const int type = train_types[b];

    // --- neighbor gather + sum over NB ---
    float v0 = 0.f, v1 = 0.f;
    const int* nn = node_neigh + ((size_t)b * TT + wv) * NBB;
#pragma unroll
    for (int n = 0; n < NBB; ++n) {
        const int idx = nn[n];
        const float* p = node_type_emb + ((size_t)idx * TT + wv) * UU;
        if (n + 1 < NBB) {
            const float* pn = node_type_emb + ((size_t)nn[n + 1] * TT + wv) * UU;
            __builtin_prefetch(pn, 0, 1);      // global_prefetch_b8
        }
        v0 += p[lane];
        v1 += p[lane + 32];
    }
    s_nte[wv][lane]      = v0;
    s_nte[wv][lane + 32] = v1;
    __syncthreads();

    // --- h[t][a] = tanh( nte[t,:] . ws1[type][:,a] ), one a per lane ---
    float h = 0.f;
    const float* w1 = ws1 + (size_t)type * UU * AA + lane;
#pragma unroll 8
    for (int u = 0; u < UU; ++u) h += s_nte[wv][u] * w1[u * AA];
    h = tanhf(h);

    // --- score[t] = sum_a h[t][a] * ws2[type][a] (wave32 shuffle reduce) ---
    float sc = h * ws2[(size_t)type * AA + lane];
#pragma unroll
    for (int off = 16; off > 0; off >>= 1) sc += __shfl_down(sc, off, 32);
    if (lane == 0) s_score[wv] = sc;
    __syncthreads();

    // --- softmax over T, agg[u] = sum_t att[t] * nte[t][u] ---
    if (tid < UU) {
        float m = s_score[0];
#pragma unroll
        for (int t = 1; t < TT; ++t) m = fmaxf(m, s_score[t]);
        float att[TT];
        float s = 0.f;
#pragma unroll
        for (int t = 0; t < TT; ++t) { att[t] = expf(s_score[t] - m); s += att[t]; }
        const float inv = 1.f / s;
        float a = 0.f;
#pragma unroll
        for (int t = 0; t < TT; ++t) a += att[t] * inv * s_nte[t][tid];
        agg_out[(size_t)b * UU + tid] = a;
    }
}

// ---------------------------------------------------------------------------
// Kernel 2: masked-dense MoE GEMM on WMMA f32 16x16x4 + bias + L2 normalize.
//   out[b,:] = normalize( node_emb[train_inputs[b],:] + A_masked[b,:] @ Wcat )
//   A_masked[b, t*64+u] = (train_types[b]==t) ? agg[b,u] : 0
// One block = 16 samples x full E=512. Wave w owns columns [w*64, w*64+64)
// as 4 accumulator tiles of 16x16. K loop: 8 type-blocks x 16 wmma K-steps.
// A pairs are hoisted into registers once; the mask is a float multiply, so
// the inner loop is pure {clause'd B loads -> 4x wmma} with no LDS / no EXEC
// manipulation.
// ---------------------------------------------------------------------------
__global__ __launch_bounds__(256) void gatne_moe_wmma(
    const float* __restrict__ node_emb,        // [N, E]
    const float* __restrict__ W,               // [T, U, E]
    const int*   __restrict__ train_inputs,    // [B]
    const int*   __restrict__ train_types,     // [B]
    const float* __restrict__ agg,             // [B, U]
    float*       __restrict__ out)             // [B, E]
{
    __shared__ float s_agg[16][UU];            // 4 KB
    __shared__ int   s_type[16];
    __shared__ float s_y[16][EE];              // 32 KB staging for row norms
    __shared__ float s_inv[16];

    const int tid  = threadIdx.x;
    const int wv   = tid >> 5;
    const int lane = tid & 31;
    const int s0   = blockIdx.x * 16;          // first sample row of this tile

    // --- async-copy agg tile (16x64 f32 = 4KB) straight into LDS -----------
    // 256 threads x 16B each, GV mode: VDST = LDS byte addr, VADDR = 64b ptr.
    {
        const float*   gsrc = agg + (size_t)s0 * UU + tid * 4;
        const uint32_t ldsa = (uint32_t)(uintptr_t)(&s_agg[0][0]) + tid * 16;
        asm volatile("global_load_async_to_lds_b128 %0, %1, off"
                     :: "v"(ldsa), "v"(gsrc) : "memory");
    }
    if (tid < 16) s_type[tid] = train_types[s0 + tid];
    asm volatile("s_wait_asynccnt 0" ::: "memory");
    __syncthreads();

    const int row   = lane & 15;   // M row for A frag / N col for B frag
    const int half  = lane >> 4;   // selects K pair {0,1} vs {2,3}
    const int nbase = wv * 64;

    // --- hoist this lane's A-row K-pairs (16 x ds_load_b64) and its type ---
    const int myType = s_type[row];
    v2f apair[16];
#pragma unroll
    for (int j = 0; j < 16; ++j) {
        const int u0 = 4 * j + 2 * half;       // even -> 8B aligned
        apair[j] = *(const v2f*)&s_agg[row][u0];
    }

    v8f acc[4] = {};               // 4 tiles of 16x16 f32

#pragma unroll
    for (int t = 0; t < TT; ++t) {
        const float msk = (myType == t) ? 1.0f : 0.0f;
        const float* Wt = W + (size_t)t * UU * EE;
#pragma unroll 4
        for (int j = 0; j < 16; ++j) {         // K-steps of 4 within type block
            const int u0 = 4 * j + 2 * half;
            const v2f a = apair[j] * msk;
#pragma unroll
            for (int nt = 0; nt < 4; ++nt) {
                const int col = nbase + nt * 16 + row;
                v2f bf;
                bf.x = Wt[(size_t)u0 * EE + col];
                bf.y = Wt[(size_t)(u0 + 1) * EE + col];
                acc[nt] = __builtin_amdgcn_wmma_f32_16x16x4_f32(
                    false, a, false, bf, (short)0, acc[nt], false, false);
            }
        }
    }

    // epilogue part 1: bias add with gathered node embedding, stage to LDS.
    // D layout: element (m,n): vgpr = m&7, lane = (m>=8)*16 + n.
    int gidx[8];
#pragma unroll
    for (int v = 0; v < 8; ++v) gidx[v] = train_inputs[s0 + v + 8 * half];
#pragma unroll
    for (int nt = 0; nt < 4; ++nt) {
#pragma unroll
        for (int v = 0; v < 8; ++v) {
            const int m = v + 8 * half;
            const int n = nbase + nt * 16 + row;
            s_y[m][n] = node_emb[(size_t)gidx[v] * EE + n] + acc[nt][v];
        }
    }
    __syncthreads();

    // epilogue part 2: per-row L2 norm (2 rows per wave), clamp at 1e-12.
    for (int m = wv; m < 16; m += 8) {
        float ss = 0.f;
#pragma unroll
        for (int e = lane; e < EE; e += 32) { const float y = s_y[m][e]; ss += y * y; }
#pragma unroll
        for (int off = 16; off > 0; off >>= 1) ss += __shfl_down(ss, off, 32);
        if (lane == 0) s_inv[m] = 1.f / fmaxf(sqrtf(ss), 1e-12f);
    }
    __syncthreads();

    // epilogue part 3: scaled, fully coalesced store of 16x512 tile.
    for (int i = tid; i < 16 * EE; i += 256) {
        const int m = i >> 9;
        const int e = i & (EE - 1);
        out[(size_t)(s0 + m) * EE + e] = s_y[m][e] * s_inv[m];
    }
}

// ---------------------------------------------------------------------------
extern "C" void kernel_launch(void* const* d_in, const int* in_sizes, int n_in,
                              void* d_out, int out_size, void* d_ws, size_t ws_size,
                              hipStream_t stream) {
    const float* node_emb     = (const float*)d_in[0];  // [N, E]
    const float* node_type_em = (const float*)d_in[1];  // [N, T, U]
    const float* W            = (const float*)d_in[2];  // [T, U, E]
    const float* ws1          = (const float*)d_in[3];  // [T, U, A]
    const float* ws2          = (const float*)d_in[4];  // [T, A, 1]
    const int*   train_inputs = (const int*)d_in[5];    // [B]
    const int*   train_types  = (const int*)d_in[6];    // [B]
    const int*   node_neigh   = (const int*)d_in[7];    // [B, T, NB]
    float*       out          = (float*)d_out;          // [B, E]

    const int B = in_sizes[5];
    float* agg = (float*)d_ws;                          // B * 64 floats (2 MB)

    gatne_gather_attn<<<B, 256, 0, stream>>>(
        node_type_em, ws1, ws2, train_types, node_neigh, agg);
    gatne_moe_wmma<<<B / 16, 256, 0, stream>>>(
        node_emb, W, train_inputs, train_types, agg, out);
}